// CTCLossLayer_68736656605679
// MI455X (gfx1250) — compile-verified
//
#include <hip/hip_runtime.h>

// CTC forward (negative log-likelihood), blank = C-1, tf.keras ctc_batch_cost semantics.
// B=128, T=1024, C=256, L=128 -> S=257 states.
// One workgroup (8 wave32s) per batch element. Alpha recurrence ping-pongs in LDS;
// softmax rows stream global->LDS via the CDNA5 async copy engine with a TRIPLE
// buffer so only ONE barrier per timestep is needed.

#define CTC_T 1024
#define CTC_C 256
#define CTC_L 128
#define CTC_S (2 * CTC_L + 1) /* 257 */

#define NEGINF (-1e30f)
#define CTC_EPS (1e-7f)

#if __has_builtin(__builtin_amdgcn_global_load_async_to_lds_b32) && \
    __has_builtin(__builtin_amdgcn_s_wait_asynccnt)
#define CTC_ASYNC 1
#else
#define CTC_ASYNC 0
#endif

typedef __attribute__((address_space(1))) int GI; // global ("__device__ int *")
typedef __attribute__((address_space(3))) int LI; // LDS    ("__shared__ int *")

__device__ __forceinline__ float lae(float a, float b) {
  // logaddexp, safe for the -1e30 "-inf" surrogate (exp underflows to 0).
  float m = fmaxf(a, b);
  float d = fabsf(a - b);
  return m + __logf(1.0f + __expf(-d));
}

__global__ __launch_bounds__(256) void ctc_loss_kernel(
    const int* __restrict__ y_true,   // [B, L] int32
    const float* __restrict__ y_pred, // [B, T, C] softmax probs
    float* __restrict__ out)          // [B, 1]
{
  __shared__ float rowbuf[3][CTC_C]; // triple-buffered y_pred rows
  __shared__ float A[2][CTC_S];      // alpha ping-pong
  __shared__ int lab[CTC_L];

  const int tid = (int)threadIdx.x; // 0..255; thread 255 also owns state 256
  const int b = (int)blockIdx.x;
  const float* src = y_pred + (size_t)b * CTC_T * CTC_C;

  if (tid < CTC_L) lab[tid] = y_true[b * CTC_L + tid];

#if CTC_ASYNC
  // Prefetch rows 0 and 1 (one async b32 per lane; 128B coalesced per wave).
  __builtin_amdgcn_global_load_async_to_lds_b32(
      (GI*)(src + tid), (LI*)&rowbuf[0][tid], 0, 0);
  __builtin_amdgcn_global_load_async_to_lds_b32(
      (GI*)(src + CTC_C + tid), (LI*)&rowbuf[1][tid], 0, 0);
#endif
  __syncthreads(); // lab[] visible; async copies in flight

  // Extended symbol + skip flag for state s = tid (thread 255 also does s=256).
  int col = CTC_C - 1; // blank for even states
  bool allow = false;  // s-2 -> s skip allowed
  if (tid & 1) {
    int li = (tid - 1) >> 1;
    col = lab[li];
    allow = (tid >= 3) && (col != lab[li - 1]); // labels never equal blank
  }
  const bool lastth = (tid == CTC_C - 1);

  // ---- t = 0: alpha0 = (s < 2) ? lp : NEG ----
#if CTC_ASYNC
  __builtin_amdgcn_s_wait_asynccnt(1); // row 0 landed (row 1 may be in flight)
#else
  rowbuf[0][tid] = src[tid];
#endif
  __syncthreads();
#if CTC_ASYNC
  __builtin_amdgcn_global_load_async_to_lds_b32(
      (GI*)(src + 2 * (size_t)CTC_C + tid), (LI*)&rowbuf[2][tid], 0, 0);
#endif
  {
    float lp = __logf(rowbuf[0][col] + CTC_EPS);
    A[0][tid] = (tid < 2) ? lp : NEGINF;
    if (lastth) A[0][CTC_S - 1] = NEGINF; // state 256
  }
  // no barrier here: next iteration's top barrier orders alpha + row buffers

  // ---- t = 1 .. T-1 : one barrier per timestep ----
#pragma unroll 3
  for (int t = 1; t < CTC_T; ++t) {
    const int cur = t % 3;       // row buffer holding row t
    const int pfb = (t + 2) % 3; // prefetch target (readers done before barrier)
    const int rb = t & 1;        // alpha buffer to write
    const int pa = rb ^ 1;       // alpha buffer from previous step

#if CTC_ASYNC
    // Invariant: exactly rows t and t+1 outstanding -> cnt<=1 means row t landed.
    __builtin_amdgcn_s_wait_asynccnt(1);
#endif
    __syncthreads(); // row t visible to all waves; alpha[pa] published; bufs reusable

#if CTC_ASYNC
    {
      int tn = t + 2;
      if (tn > CTC_T - 1) tn = CTC_T - 1; // clamped junk keeps 2-outstanding invariant
      __builtin_amdgcn_global_load_async_to_lds_b32(
          (GI*)(src + (size_t)tn * CTC_C + tid), (LI*)&rowbuf[pfb][tid], 0, 0);
    }
#else
    rowbuf[cur][tid] = src[(size_t)t * CTC_C + tid];
    __syncthreads();
#endif

    {
      float a0 = A[pa][tid];
      float a1 = (tid >= 1) ? A[pa][tid - 1] : NEGINF;
      float a = lae(a0, a1);
      if (allow) a = lae(a, A[pa][tid - 2]); // allow implies tid >= 3
      float lp = __logf(rowbuf[cur][col] + CTC_EPS);
      A[rb][tid] = a + lp;
      if (lastth) { // state 256: blank, transitions from s=255,256 only
        float b0 = A[pa][CTC_S - 1];
        float b1 = A[pa][CTC_S - 2];
        float lpb = __logf(rowbuf[cur][CTC_C - 1] + CTC_EPS);
        A[rb][CTC_S - 1] = lae(b0, b1) + lpb;
      }
    }
  }

  __syncthreads();
  if (tid == 0) {
    const int pf = (CTC_T - 1) & 1;
    out[b] = -lae(A[pf][CTC_S - 1], A[pf][CTC_S - 2]);
  }
}

extern "C" void kernel_launch(void* const* d_in, const int* in_sizes, int n_in,
                              void* d_out, int out_size, void* d_ws, size_t ws_size,
                              hipStream_t stream) {
  (void)n_in; (void)d_ws; (void)ws_size; (void)in_sizes;
  const int* y_true = (const int*)d_in[0];
  const float* y_pred = (const float*)d_in[1];
  float* out = (float*)d_out;
  const int B = out_size; // [B, 1] -> B elements
  ctc_loss_kernel<<<dim3(B), dim3(256), 0, stream>>>(y_true, y_pred, out);
}